// BSplineField3d_13821204759219
// MI455X (gfx1250) — compile-verified
//
#include <hip/hip_runtime.h>

// BSpline 3D displacement field evaluation for gfx1250 (MI455X).
// Gather-bound kernel: phi (3 MB) stays L2/WGP$-resident via RT policy while
// the 96 MB of stream-once point/output data uses non-temporal hints.
// Separable tensor-product contraction: contract z (contiguous float3 taps)
// first, then y, then x -> 252 FMA/point instead of 448.

#define GN 64                 // grid size per dimension
#define GDIM 3                // field dimension
#define SLICE (GN * GN * GDIM)   // stride for ix
#define ROW   (GN * GDIM)        // stride for iy

__device__ __forceinline__ void bspline_weights(float u, float w[4]) {
    const float k = 1.0f / 6.0f;
    const float s  = 1.0f - u;
    const float u2 = u * u;
    const float u3 = u2 * u;
    w[0] = s * s * s * k;
    w[1] = (3.0f * u3 - 6.0f * u2 + 4.0f) * k;
    w[2] = (-3.0f * u3 + 3.0f * u2 + 3.0f * u + 1.0f) * k;
    w[3] = u3 * k;
}

__global__ __launch_bounds__(256) void bspline_field3d_kernel(
    const float* __restrict__ xs,
    const float* __restrict__ ys,
    const float* __restrict__ zs,
    const float* __restrict__ phi,   // [64][64][64][3]
    float* __restrict__ out,         // [N][3]
    int n)
{
    const float inv_d = 30.5f;       // 1/dx = (64-3)/2
    const int stride = blockDim.x * gridDim.x;

    for (int i = blockIdx.x * blockDim.x + threadIdx.x; i < n; i += stride) {
        // Prefetch next grid-stride chunk of the streaming inputs
        // (emits global_prefetch_b8 on gfx1250).
        if (i + stride < n) {
            __builtin_prefetch(&xs[i + stride], 0, 0);
            __builtin_prefetch(&ys[i + stride], 0, 0);
            __builtin_prefetch(&zs[i + stride], 0, 0);
        }

        // Stream-once inputs: non-temporal so phi keeps the cache.
        float u = (__builtin_nontemporal_load(&xs[i]) + 1.0f) * inv_d;
        float v = (__builtin_nontemporal_load(&ys[i]) + 1.0f) * inv_d;
        float w = (__builtin_nontemporal_load(&zs[i]) + 1.0f) * inv_d;

        const float fu = floorf(u);
        const float fv = floorf(v);
        const float fw = floorf(w);
        const int ix = (int)fu;
        const int iy = (int)fv;
        const int iz = (int)fw;
        u -= fu; v -= fv; w -= fw;

        float wu[4], wv[4], ww[4];
        bspline_weights(u, wu);
        bspline_weights(v, wv);
        bspline_weights(w, ww);

        // Clamped tap indices (pre-scaled to element offsets).
        int cx[4], cy[4], cz[4];
#pragma unroll
        for (int t = 0; t < 4; ++t) {
            int a = ix + t; a = a < 0 ? 0 : (a > GN - 1 ? GN - 1 : a);
            int b = iy + t; b = b < 0 ? 0 : (b > GN - 1 ? GN - 1 : b);
            int c = iz + t; c = c < 0 ? 0 : (c > GN - 1 ? GN - 1 : c);
            cx[t] = a * SLICE;
            cy[t] = b * ROW;
            cz[t] = c * GDIM;
        }

        float Tx = 0.0f, Ty = 0.0f, Tz = 0.0f;
#pragma unroll
        for (int l = 0; l < 4; ++l) {
            const float* __restrict__ pl = phi + cx[l];
            float ax = 0.0f, ay = 0.0f, az = 0.0f;
#pragma unroll
            for (int m = 0; m < 4; ++m) {
                const float* __restrict__ pm = pl + cy[m];
                float bx = 0.0f, by = 0.0f, bz = 0.0f;
#pragma unroll
                for (int q = 0; q < 4; ++q) {
                    const float* __restrict__ p = pm + cz[q];
                    const float wq = ww[q];
                    bx = fmaf(wq, p[0], bx);
                    by = fmaf(wq, p[1], by);
                    bz = fmaf(wq, p[2], bz);
                }
                ax = fmaf(wv[m], bx, ax);
                ay = fmaf(wv[m], by, ay);
                az = fmaf(wv[m], bz, az);
            }
            Tx = fmaf(wu[l], ax, Tx);
            Ty = fmaf(wu[l], ay, Ty);
            Tz = fmaf(wu[l], az, Tz);
        }

        // Streamed output: non-temporal store (write-once, never re-read).
        __builtin_nontemporal_store(Tx, &out[3 * i + 0]);
        __builtin_nontemporal_store(Ty, &out[3 * i + 1]);
        __builtin_nontemporal_store(Tz, &out[3 * i + 2]);
    }
}

extern "C" void kernel_launch(void* const* d_in, const int* in_sizes, int n_in,
                              void* d_out, int out_size, void* d_ws, size_t ws_size,
                              hipStream_t stream) {
    const float* xs  = (const float*)d_in[0];
    const float* ys  = (const float*)d_in[1];
    const float* zs  = (const float*)d_in[2];
    const float* phi = (const float*)d_in[3];
    float* out = (float*)d_out;
    const int n = in_sizes[0];

    const int block = 256;                       // 8 wave32s per block
    int grid = (n + block - 1) / block;
    if (grid > 4096) grid = 4096;                // grid-stride: ~4 pts/thread at 4M
    bspline_field3d_kernel<<<grid, block, 0, stream>>>(xs, ys, zs, phi, out, n);
}